// AttentionHeadVDP_85255100826166
// MI455X (gfx1250) — compile-verified
//
#include <hip/hip_runtime.h>
#include <hip/hip_bf16.h>

typedef __attribute__((ext_vector_type(2))) float v2f;
typedef __attribute__((ext_vector_type(8))) float v8f;

#define D_  1024
#define S_  1024
#define B_  4
#define H_  16
#define DH_ 64

// D = A(16x4 f32) * B(4x16 f32) + C(16x16 f32)
static __device__ __forceinline__ v8f wmma4(v2f a, v2f b, v8f c) {
  return __builtin_amdgcn_wmma_f32_16x16x4_f32(false, a, false, b, (short)0, c,
                                               false, false);
}

static __device__ __forceinline__ v8f zero8() {
  v8f z = {0.f, 0.f, 0.f, 0.f, 0.f, 0.f, 0.f, 0.f};
  return z;
}

// Direct global -> LDS async copy of 16 bytes per lane (ASYNCcnt-tracked).
static __device__ __forceinline__ void async_copy16(const float* lds_dst,
                                                    const float* gsrc) {
  const unsigned lds_off = (unsigned)(size_t)(const void*)lds_dst;
  asm volatile("global_load_async_to_lds_b128 %0, %1, off"
               :
               : "v"(lds_off), "v"(gsrc)
               : "memory");
}
static __device__ __forceinline__ void async_drain() {
  asm volatile("s_wait_asynccnt 0" ::: "memory");
}

static __device__ __forceinline__ float redmax16(float v) {
  v = fmaxf(v, __shfl_xor(v, 1));
  v = fmaxf(v, __shfl_xor(v, 2));
  v = fmaxf(v, __shfl_xor(v, 4));
  v = fmaxf(v, __shfl_xor(v, 8));
  return v;
}
static __device__ __forceinline__ float redsum16(float v) {
  v += __shfl_xor(v, 1);
  v += __shfl_xor(v, 2);
  v += __shfl_xor(v, 4);
  v += __shfl_xor(v, 8);
  return v;
}

// ---------------------------------------------------------------------------
// Kernel 1: VDP linear projection, head-split output [B,H,S,dh]
//   mu  = x @ w.T
//   var = var_x @ (w^2 + var_w).T + x^2 @ var_w.T
// grid (D/64, BS/64), block 128 (4 waves; wave = 16 output rows x 64 cols)
// ---------------------------------------------------------------------------
#define PSTR 36  // LDS row stride (floats) for 32-wide K tiles (16B-aligned rows)

__global__ __launch_bounds__(128) void vdp_proj_kernel(
    const float* __restrict__ x, const float* __restrict__ vx,
    const float* __restrict__ w, const float* __restrict__ vw,
    float* __restrict__ omu, float* __restrict__ ovar) {
  const int tid  = threadIdx.x;
  const int lane = tid & 31;
  const int wv   = tid >> 5;
  const int n0   = blockIdx.x * 64;  // output column tile
  const int m0   = blockIdx.y * 64;  // row tile over B*S

  __shared__ float xs[64 * PSTR];
  __shared__ float vxs[64 * PSTR];
  __shared__ float wts[64 * PSTR];
  __shared__ float vwts[64 * PSTR];

  v8f cmu[4], cvar[4];
#pragma unroll
  for (int i = 0; i < 4; i++) { cmu[i] = zero8(); cvar[i] = zero8(); }

  const int hlane = (lane >> 4);  // 0 or 1
  const int llane = (lane & 15);

  for (int kk = 0; kk < D_; kk += 32) {
    __syncthreads();
#pragma unroll
    for (int i0 = 0; i0 < 512; i0 += 128) {  // 64 rows x 8 float4
      const int i = i0 + tid;
      const int r = i >> 3;
      const int c = (i & 7) << 2;
      const size_t gx = (size_t)(m0 + r) * D_ + kk + c;
      const size_t gw = (size_t)(n0 + r) * D_ + kk + c;
      async_copy16(&xs[r * PSTR + c], x + gx);
      async_copy16(&vxs[r * PSTR + c], vx + gx);
      async_copy16(&wts[r * PSTR + c], w + gw);
      async_copy16(&vwts[r * PSTR + c], vw + gw);
    }
    async_drain();
    __syncthreads();

    const int mA = wv * 16 + llane;
#pragma unroll
    for (int kc = 0; kc < 8; kc++) {
      const int ka = kc * 4 + (hlane << 1);
      v2f ax  = {xs[mA * PSTR + ka], xs[mA * PSTR + ka + 1]};
      v2f av  = {vxs[mA * PSTR + ka], vxs[mA * PSTR + ka + 1]};
      v2f ax2 = ax * ax;
#pragma unroll
      for (int ct = 0; ct < 4; ct++) {
        const int bn = ct * 16 + llane;
        v2f bw = {wts[bn * PSTR + ka], wts[bn * PSTR + ka + 1]};
        v2f bv = {vwts[bn * PSTR + ka], vwts[bn * PSTR + ka + 1]};
        v2f bc = bw * bw + bv;
        cmu[ct]  = wmma4(ax, bw, cmu[ct]);
        cvar[ct] = wmma4(av, bc, cvar[ct]);
        cvar[ct] = wmma4(ax2, bv, cvar[ct]);
      }
    }
  }

  // epilogue: head-split store [B,H,S,dh]
#pragma unroll
  for (int ct = 0; ct < 4; ct++) {
    const int n = n0 + ct * 16 + llane;
    const int h = n >> 6;
    const int c = n & 63;
#pragma unroll
    for (int r = 0; r < 8; r++) {
      const int rg = m0 + wv * 16 + r + (hlane << 3);
      const int b = rg >> 10;
      const int s = rg & 1023;
      const size_t o = ((size_t)(b * H_ + h) * S_ + s) * DH_ + c;
      omu[o]  = cmu[ct][r];
      ovar[o] = cvar[ct][r];
    }
  }
}

// ---------------------------------------------------------------------------
// Kernel 2: flash-style VDP attention over head-split q/k/v, with residual.
// grid (S/64, B*H), block 128 (4 waves; wave = 16 query rows).
// ---------------------------------------------------------------------------
#define KSTR 68  // LDS row stride (floats) for 64-wide K/V tiles (16B-aligned)

__global__ __launch_bounds__(128) void vdp_attn_kernel(
    const float* __restrict__ x, const float* __restrict__ vxin,
    const float* __restrict__ qmu, const float* __restrict__ qvar,
    const float* __restrict__ kmu, const float* __restrict__ kvar,
    const float* __restrict__ vmu, const float* __restrict__ vvar,
    float* __restrict__ out_mu, float* __restrict__ out_var) {
  const int tid   = threadIdx.x;
  const int lane  = tid & 31;
  const int wv    = tid >> 5;
  const int qbase = blockIdx.x * 64;
  const int bh    = blockIdx.y;
  const int b     = bh >> 4;
  const int h     = bh & 15;
  const int hlane = (lane >> 4);
  const int llane = (lane & 15);

  __shared__ float kms[32 * KSTR];
  __shared__ float kvs[32 * KSTR];
  __shared__ float vms[32 * KSTR];
  __shared__ float vvs[32 * KSTR];
  __shared__ float pscr[4][16 * PSTR];
  __shared__ float vascr[4][16 * PSTR];

  // per-wave Q fragments (16 rows x 64 dh)
  const int qrow = bh * S_ + qbase + wv * 16 + llane;
  const float* qmp = qmu + (size_t)qrow * DH_;
  const float* qvp = qvar + (size_t)qrow * DH_;
  v2f aqm[16], aqv[16];
#pragma unroll
  for (int kc = 0; kc < 16; kc++) {
    const int ka = kc * 4 + (hlane << 1);
    aqm[kc] = (v2f){qmp[ka], qmp[ka + 1]};
    aqv[kc] = (v2f){qvp[ka], qvp[ka + 1]};
  }

  float mrow[8], Zr[8], Tr[8];
#pragma unroll
  for (int r = 0; r < 8; r++) { mrow[r] = -1e30f; Zr[r] = 0.f; Tr[r] = 0.f; }

  const size_t kvbase = (size_t)bh * S_ * DH_;
  const float inv_rd  = 0.03125f;      // 1/sqrt(D)
  const float inv_d   = 9.765625e-4f;  // 1/D

  // ---------------- pass 1: softmax statistics (m, Z, T) ----------------
  for (int kt = 0; kt < 32; kt++) {
    __syncthreads();
    const size_t tb = kvbase + (size_t)kt * 32 * DH_;
#pragma unroll
    for (int i0 = 0; i0 < 512; i0 += 128) {  // 32 rows x 16 float4, contiguous
      const int i = i0 + tid;
      const int r = i >> 4;
      const int c = (i & 15) << 2;
      async_copy16(&kms[r * KSTR + c], kmu + tb + (size_t)i * 4);
      async_copy16(&kvs[r * KSTR + c], kvar + tb + (size_t)i * 4);
    }
    async_drain();
    __syncthreads();

    for (int dt = 0; dt < 2; dt++) {
      v8f smu = zero8(), svar = zero8();
      const int bn = dt * 16 + llane;
#pragma unroll
      for (int kc = 0; kc < 16; kc++) {
        const int ka = kc * 4 + (hlane << 1);
        v2f bm = {kms[bn * KSTR + ka], kms[bn * KSTR + ka + 1]};
        v2f bv = {kvs[bn * KSTR + ka], kvs[bn * KSTR + ka + 1]};
        v2f ax = aqm[kc];
        v2f av = aqv[kc];
        smu  = wmma4(ax, bm, smu);
        svar = wmma4(ax * ax, bv, svar);
        svar = wmma4(av, bm * bm + bv, svar);
      }
#pragma unroll
      for (int r = 0; r < 8; r++) {
        const float ms = smu[r] * inv_rd;
        const float vs = fmaxf(svar[r], 1e-3f) * inv_d;  // clip then /RD^2
        const float tm = redmax16(ms);
        const float mnew = fmaxf(mrow[r], tm);
        const float e  = __expf(ms - mnew);
        const float es = redsum16(e);
        const float ts = redsum16(e * e * vs);
        const float corr = __expf(mrow[r] - mnew);
        Zr[r] = Zr[r] * corr + es;
        Tr[r] = Tr[r] * corr * corr + ts;
        mrow[r] = mnew;
      }
    }
  }
#pragma unroll
  for (int r = 0; r < 8; r++) {
    const float iz = 1.0f / Zr[r];
    Tr[r] = Tr[r] * iz * iz;  // s = sum p^2 var
    Zr[r] = iz;               // 1/Z
  }

  v8f omu[4], ovar[4];
#pragma unroll
  for (int i = 0; i < 4; i++) { omu[i] = zero8(); ovar[i] = zero8(); }

  // ---------------- pass 2: recompute scores, P*V with variance ----------
  for (int kt = 0; kt < 32; kt++) {
    __syncthreads();
    const size_t tb = kvbase + (size_t)kt * 32 * DH_;
#pragma unroll
    for (int i0 = 0; i0 < 512; i0 += 128) {
      const int i = i0 + tid;
      const int r = i >> 4;
      const int c = (i & 15) << 2;
      async_copy16(&kms[r * KSTR + c], kmu + tb + (size_t)i * 4);
      async_copy16(&kvs[r * KSTR + c], kvar + tb + (size_t)i * 4);
      async_copy16(&vms[r * KSTR + c], vmu + tb + (size_t)i * 4);
      async_copy16(&vvs[r * KSTR + c], vvar + tb + (size_t)i * 4);
    }
    async_drain();
    __syncthreads();

    for (int dt = 0; dt < 2; dt++) {
      v8f smu = zero8(), svar = zero8();
      const int bn = dt * 16 + llane;
#pragma unroll
      for (int kc = 0; kc < 16; kc++) {
        const int ka = kc * 4 + (hlane << 1);
        v2f bm = {kms[bn * KSTR + ka], kms[bn * KSTR + ka + 1]};
        v2f bv = {kvs[bn * KSTR + ka], kvs[bn * KSTR + ka + 1]};
        v2f ax = aqm[kc];
        v2f av = aqv[kc];
        smu  = wmma4(ax, bm, smu);
        svar = wmma4(ax * ax, bv, svar);
        svar = wmma4(av, bm * bm + bv, svar);
      }
#pragma unroll
      for (int r = 0; r < 8; r++) {
        const float ms = smu[r] * inv_rd;
        const float vs = fmaxf(svar[r], 1e-3f) * inv_d;
        const float p  = __expf(ms - mrow[r]) * Zr[r];
        float va = p * p * (Tr[r] + (1.f - 2.f * p) * vs);
        va = fmaxf(va, 1e-3f);  // softmax_vdp clip
        const int row = r + (hlane << 3);
        pscr[wv][row * PSTR + dt * 16 + llane]  = p;
        vascr[wv][row * PSTR + dt * 16 + llane] = va;
      }
    }
    __syncthreads();

#pragma unroll
    for (int kc = 0; kc < 8; kc++) {
      const int ka = kc * 4 + (hlane << 1);
      v2f ap = {pscr[wv][llane * PSTR + ka], pscr[wv][llane * PSTR + ka + 1]};
      v2f aa = {vascr[wv][llane * PSTR + ka], vascr[wv][llane * PSTR + ka + 1]};
#pragma unroll
      for (int ct = 0; ct < 4; ct++) {
        const int col = ct * 16 + llane;
        v2f bm = {vms[ka * KSTR + col], vms[(ka + 1) * KSTR + col]};
        v2f bv = {vvs[ka * KSTR + col], vvs[(ka + 1) * KSTR + col]};
        omu[ct]  = wmma4(ap, bm, omu[ct]);
        ovar[ct] = wmma4(ap * ap, bv, ovar[ct]);
        ovar[ct] = wmma4(aa, bm * bm + bv, ovar[ct]);
      }
    }
  }

  // epilogue: residual, final variance clip, merge heads
#pragma unroll
  for (int ct = 0; ct < 4; ct++) {
    const int d = h * 64 + ct * 16 + llane;
#pragma unroll
    for (int r = 0; r < 8; r++) {
      const int srow = qbase + wv * 16 + r + (hlane << 3);
      const size_t o = ((size_t)(b * S_) + srow) * D_ + d;
      out_mu[o]  = x[o] + omu[ct][r];
      out_var[o] = vxin[o] + fmaxf(ovar[ct][r], 1e-3f);
    }
  }
}

// ---------------------------------------------------------------------------
extern "C" void kernel_launch(void* const* d_in, const int* in_sizes, int n_in,
                              void* d_out, int out_size, void* d_ws,
                              size_t ws_size, hipStream_t stream) {
  (void)in_sizes; (void)n_in; (void)out_size; (void)ws_size;
  const float* x  = (const float*)d_in[0];
  const float* vx = (const float*)d_in[1];

  float* ws = (float*)d_ws;
  const size_t SEG = (size_t)B_ * H_ * S_ * DH_;  // 4M floats per tensor

  // projections: z=0 -> q, 1 -> k, 2 -> v
  for (int z = 0; z < 3; z++) {
    const float* w  = (const float*)d_in[2 + 2 * z];
    const float* vw = (const float*)d_in[3 + 2 * z];
    vdp_proj_kernel<<<dim3(D_ / 64, (B_ * S_) / 64), 128, 0, stream>>>(
        x, vx, w, vw, ws + (size_t)z * 2 * SEG, ws + (size_t)z * 2 * SEG + SEG);
  }

  float* om = (float*)d_out;
  float* ov = om + (size_t)B_ * S_ * D_;
  vdp_attn_kernel<<<dim3(S_ / 64, B_ * H_), 128, 0, stream>>>(
      x, vx, ws + 0 * SEG, ws + 1 * SEG, ws + 2 * SEG, ws + 3 * SEG,
      ws + 4 * SEG, ws + 5 * SEG, om, ov);
}